// CrossMultiAttention_updateSP_7146825581105
// MI455X (gfx1250) — compile-verified
//
#include <hip/hip_runtime.h>

// CrossMultiAttention updateSP for MI455X (gfx1250, wave32).
// Dominant GEMM (raw @ Wfold.T, 8.6 GFLOP) + Q projection run on
// V_WMMA_F32_16X16X4_F32 to keep full fp32 precision. Everything else is
// bandwidth-bound (raw = 67 MB/pass @ 23.3 TB/s) with L2-resident segment
// softmax state (4096*8 max/denom, 2 MB context) updated via atomics.

#define S_    4096
#define N_    262144
#define H_    8
#define E_    128
#define D_    16
#define CKV_  64
#define EPS_  1e-5f

typedef float v2f __attribute__((ext_vector_type(2)));
typedef float v4f __attribute__((ext_vector_type(4)));
typedef float v8f __attribute__((ext_vector_type(8)));

// Order-preserving float<->uint encoding for segment max via atomicMax.
// memset(0) is the identity (maps below every real float's encoding).
__device__ __forceinline__ unsigned enc_f32(float f) {
  unsigned u = __float_as_uint(f);
  return (u & 0x80000000u) ? ~u : (u | 0x80000000u);
}
__device__ __forceinline__ float dec_f32(unsigned e) {
  return (e & 0x80000000u) ? __uint_as_float(e & 0x7fffffffu)
                           : __uint_as_float(~e);
}

// ---------------------------------------------------------------- K0: fold
// Wfold[o, j] = Wkv[o, j] + Wkv[o, j+64]   (concat([x,x]) @ Wkv.T trick)
__global__ void k_fold_wkv(const float* __restrict__ Wkv,
                           float* __restrict__ Wfold) {
  int i = blockIdx.x * blockDim.x + threadIdx.x;   // 0 .. 256*64-1
  int o = i >> 6, j = i & 63;
  Wfold[i] = Wkv[o * (2 * CKV_) + j] + Wkv[o * (2 * CKV_) + CKV_ + j];
}

// ---------------------------------------------------------------- K1: Q GEMM
// Q = sp_feat(4096x128) @ Wq.T(128x128), fp32 WMMA, one wave per 16x16 tile.
__global__ void k_qproj(const float* __restrict__ sp,
                        const float* __restrict__ Wq,
                        float* __restrict__ Q) {
  const int tm   = blockIdx.x * 16;
  const int tn   = blockIdx.y * 16;
  const int lane = threadIdx.x;
  const int half = lane >> 4;
  const int l    = lane & 15;
  // A lane pair: sp[tm+l][k + 2*half .. +1]; B lane pair: Wq[tn+l][k + 2*half .. +1]
  const float* arow = sp + (size_t)(tm + l) * E_ + 2 * half;
  const float* brow = Wq + (size_t)(tn + l) * E_ + 2 * half;
  v8f acc = {0.f, 0.f, 0.f, 0.f, 0.f, 0.f, 0.f, 0.f};
#pragma unroll
  for (int k = 0; k < E_; k += 4) {
    v2f a = *(const v2f*)(arow + k);
    v2f b = *(const v2f*)(brow + k);
    acc = __builtin_amdgcn_wmma_f32_16x16x4_f32(false, a, false, b,
                                                (short)0, acc, false, false);
  }
#pragma unroll
  for (int r = 0; r < 8; ++r)
    Q[(size_t)(tm + r + 8 * half) * E_ + tn + l] = acc[r];
}

// ------------------------------------------------- K2: K projection + scores
// One wave per 16-point block. k_tile(16x128) via 8 head-tiles of WMMA
// (A block cached in 16 v2f regs, reused across heads), then per-(p,h)
// score = 0.25 * <Q[seg[p],h,:], k[p,h,:]>, spill to ws, segment max via
// encoded atomicMax, and attn_scores output = mean over heads.
__global__ void k_kproj_scores(const float* __restrict__ raw,
                               const float* __restrict__ Wfold,
                               const float* __restrict__ Q,
                               const int* __restrict__ assign,
                               float* __restrict__ scores,
                               unsigned* __restrict__ segmax,
                               float* __restrict__ attn_out) {
  __shared__ float kt[16 * 128];
  __shared__ float sc[16 * 8];
  const int p0   = blockIdx.x * 16;
  const int lane = threadIdx.x;
  const int half = lane >> 4;
  const int l    = lane & 15;

  const float* arow = raw + (size_t)(p0 + l) * CKV_ + 2 * half;
  v2f areg[16];
#pragma unroll
  for (int k = 0; k < 16; ++k) areg[k] = *(const v2f*)(arow + 4 * k);

#pragma unroll
  for (int h = 0; h < H_; ++h) {
    const float* brow = Wfold + (size_t)(h * 16 + l) * CKV_ + 2 * half;
    v8f acc = {0.f, 0.f, 0.f, 0.f, 0.f, 0.f, 0.f, 0.f};
#pragma unroll
    for (int k = 0; k < 16; ++k) {
      v2f b = *(const v2f*)(brow + 4 * k);
      acc = __builtin_amdgcn_wmma_f32_16x16x4_f32(false, areg[k], false, b,
                                                  (short)0, acc, false, false);
    }
#pragma unroll
    for (int r = 0; r < 8; ++r)
      kt[(r + 8 * half) * 128 + h * 16 + l] = acc[r];
  }
  __syncthreads();

#pragma unroll
  for (int j = 0; j < 4; ++j) {
    int q = lane * 4 + j;           // 0..127 (point,head) pairs
    int p = q >> 3;
    int h = q & 7;
    int seg = assign[p0 + p];
    const v4f* qv = (const v4f*)(Q + (size_t)seg * E_ + h * 16);
    const v4f* kv = (const v4f*)(&kt[p * 128 + h * 16]);
    float s = 0.f;
#pragma unroll
    for (int t = 0; t < 4; ++t) {
      v4f qq = qv[t], kk = kv[t];
      s += qq.x * kk.x + qq.y * kk.y + qq.z * kk.z + qq.w * kk.w;
    }
    s *= 0.25f;                     // D^-0.5
    scores[(size_t)(p0 + p) * H_ + h] = s;
    atomicMax(&segmax[seg * H_ + h], enc_f32(s));
    sc[p * 8 + h] = s;
  }
  __syncthreads();
  if (lane < 16) {
    float m = 0.f;
#pragma unroll
    for (int h = 0; h < H_; ++h) m += sc[lane * 8 + h];
    attn_out[p0 + lane] = m * (1.f / H_);
  }
}

// --------------------------------------------- K3: V projection + aggregate
// Same tile structure (Wfold rows 128..255). w = exp(s - segmax);
// atomicAdd denom and w*v into the L2-resident context buffer.
__global__ void k_vproj_ctx(const float* __restrict__ raw,
                            const float* __restrict__ Wfold,
                            const float* __restrict__ scores,
                            const unsigned* __restrict__ segmax,
                            const int* __restrict__ assign,
                            float* __restrict__ denom,
                            float* __restrict__ ctx) {
  __shared__ float vt[16 * 128];
  const int p0   = blockIdx.x * 16;
  const int lane = threadIdx.x;
  const int half = lane >> 4;
  const int l    = lane & 15;

  const float* arow = raw + (size_t)(p0 + l) * CKV_ + 2 * half;
  v2f areg[16];
#pragma unroll
  for (int k = 0; k < 16; ++k) areg[k] = *(const v2f*)(arow + 4 * k);

#pragma unroll
  for (int h = 0; h < H_; ++h) {
    const float* brow = Wfold + (size_t)(E_ + h * 16 + l) * CKV_ + 2 * half;
    v8f acc = {0.f, 0.f, 0.f, 0.f, 0.f, 0.f, 0.f, 0.f};
#pragma unroll
    for (int k = 0; k < 16; ++k) {
      v2f b = *(const v2f*)(brow + 4 * k);
      acc = __builtin_amdgcn_wmma_f32_16x16x4_f32(false, areg[k], false, b,
                                                  (short)0, acc, false, false);
    }
#pragma unroll
    for (int r = 0; r < 8; ++r)
      vt[(r + 8 * half) * 128 + h * 16 + l] = acc[r];
  }
  __syncthreads();

#pragma unroll
  for (int j = 0; j < 4; ++j) {
    int q = lane * 4 + j;
    int p = q >> 3;
    int h = q & 7;
    int seg = assign[p0 + p];
    float s = scores[(size_t)(p0 + p) * H_ + h];
    float m = dec_f32(segmax[seg * H_ + h]);
    float w = __expf(s - m);
    atomicAdd(&denom[seg * H_ + h], w);
    const float* vrow = &vt[p * 128 + h * 16];
    float* crow = ctx + (size_t)seg * E_ + h * 16;
#pragma unroll
    for (int d = 0; d < D_; ++d) atomicAdd(&crow[d], w * vrow[d]);
  }
}

// --------------------------------------------- K4: residual + LayerNorm
// One wave per superpoint row; 4 elems/lane; wave32 shfl_xor reductions.
__global__ void k_finalize(const float* __restrict__ sp,
                           const float* __restrict__ ctx,
                           const float* __restrict__ denom,
                           const float* __restrict__ lnw,
                           const float* __restrict__ lnb,
                           float* __restrict__ out) {
  const int row  = blockIdx.x;
  const int lane = threadIdx.x;
  const int e0   = lane * 4;                 // elems e0..e0+3, one head each
  v4f c = *(const v4f*)(ctx + (size_t)row * E_ + e0);
  v4f s = *(const v4f*)(sp + (size_t)row * E_ + e0);
  float den = denom[row * H_ + (e0 >> 4)];
  float inv = (den > 0.f) ? (1.f / den) : 0.f;   // empty segment -> context 0
  float x[4] = {s.x + c.x * inv, s.y + c.y * inv,
                s.z + c.z * inv, s.w + c.w * inv};
  float sum = x[0] + x[1] + x[2] + x[3];
#pragma unroll
  for (int m = 16; m >= 1; m >>= 1) sum += __shfl_xor(sum, m, 32);
  float mu = sum * (1.f / E_);
  float vs = 0.f;
#pragma unroll
  for (int j = 0; j < 4; ++j) { float d = x[j] - mu; vs += d * d; }
#pragma unroll
  for (int m = 16; m >= 1; m >>= 1) vs += __shfl_xor(vs, m, 32);
  float rs = rsqrtf(vs * (1.f / E_) + EPS_);
  v4f w = *(const v4f*)(lnw + e0);
  v4f b = *(const v4f*)(lnb + e0);
  v4f y;
  y.x = (x[0] - mu) * rs * w.x + b.x;
  y.y = (x[1] - mu) * rs * w.y + b.y;
  y.z = (x[2] - mu) * rs * w.z + b.z;
  y.w = (x[3] - mu) * rs * w.w + b.w;
  *(v4f*)(out + (size_t)row * E_ + e0) = y;
}

extern "C" void kernel_launch(void* const* d_in, const int* in_sizes, int n_in,
                              void* d_out, int out_size, void* d_ws,
                              size_t ws_size, hipStream_t stream) {
  (void)in_sizes; (void)n_in; (void)out_size; (void)ws_size;
  const float* sp  = (const float*)d_in[0];
  const float* raw = (const float*)d_in[1];
  const int*   asg = (const int*)d_in[2];
  const float* Wq  = (const float*)d_in[3];
  const float* Wkv = (const float*)d_in[4];
  const float* lnw = (const float*)d_in[5];
  const float* lnb = (const float*)d_in[6];

  float* out      = (float*)d_out;               // updated_sp_feat (S*E)
  float* out_attn = out + (size_t)S_ * E_;       // attn_scores (N)

  // workspace layout (~12.9 MB)
  float*    ws     = (float*)d_ws;
  float*    Wfold  = ws;                                   // 256*64
  float*    Q      = Wfold + 256 * CKV_;                   // S*E
  float*    scores = Q + (size_t)S_ * E_;                  // N*H
  unsigned* segmax = (unsigned*)(scores + (size_t)N_ * H_);// S*H
  float*    denom  = (float*)(segmax + S_ * H_);           // S*H
  float*    ctx    = denom + S_ * H_;                      // S*E

  hipMemsetAsync(segmax, 0, (size_t)S_ * H_ * sizeof(unsigned), stream);
  hipMemsetAsync(denom, 0, (size_t)S_ * H_ * sizeof(float), stream);
  hipMemsetAsync(ctx, 0, (size_t)S_ * E_ * sizeof(float), stream);

  k_fold_wkv<<<64, 256, 0, stream>>>(Wkv, Wfold);
  k_qproj<<<dim3(S_ / 16, E_ / 16), 32, 0, stream>>>(sp, Wq, Q);
  k_kproj_scores<<<N_ / 16, 32, 0, stream>>>(raw, Wfold, Q, asg, scores,
                                             segmax, out_attn);
  k_vproj_ctx<<<N_ / 16, 32, 0, stream>>>(raw, Wfold, scores, segmax, asg,
                                          denom, ctx);
  k_finalize<<<S_, 32, 0, stream>>>(sp, ctx, denom, lnw, lnb, out);
}